// Attn_27419071218476
// MI455X (gfx1250) — compile-verified
//
#include <hip/hip_runtime.h>
#include <math.h>

// ---------------------------------------------------------------------------
// Problem constants (from the reference): S=2048, B=64, H=1024
// hidden: [1,B,H] f32; encoder_outputs: [S,B,H] f32; out: [B,1,S] f32
// ---------------------------------------------------------------------------
#define S_DIM 2048
#define B_DIM 64
#define H_DIM 1024
#define TILES_PER_B (S_DIM / 16)     // 128 s-tiles of 16 per batch
#define WAVES_PER_BLOCK 8            // 256 threads = 8 wave32

typedef __attribute__((ext_vector_type(2))) float v2f;
typedef __attribute__((ext_vector_type(4))) float v4f;
typedef __attribute__((ext_vector_type(8))) float v8f;

// ---------------------------------------------------------------------------
// Kernel 1: scores[b,s] = dot(hidden[b,:], enc[s,b,:]) via V_WMMA_F32_16X16X4_F32
//
// Per wave: one (b, 16-wide s tile). K-loop over H in steps of 4.
//   A (16x4, f32): hidden[b, h..h+3] broadcast to all 16 M-rows.
//       ISA layout: lanes 0-15 hold {K0,K1} in A.xy, lanes 16-31 hold {K2,K3}.
//   B (4x16, f32): column n = encoder row s0+n.
//       ISA layout (row-striped): VGPR0 = rows K0(l0-15)/K1(l16-31),
//                                 VGPR1 = rows K2(l0-15)/K3(l16-31).
//       Each lane NT-loads float4 enc[s0+laneN, b, h..h+3] and selects 2 of 4.
//   D rows are identical (A rows identical); column n holds score[s0+n].
// ---------------------------------------------------------------------------
__global__ __launch_bounds__(256) void attn_scores_wmma(
    const float* __restrict__ hidden,   // [B,H]
    const float* __restrict__ enc,      // [S,B,H]
    float* __restrict__ scores)         // [B,S]  (== d_out)
{
    const int lane  = threadIdx.x & 31;
    const int wave  = threadIdx.x >> 5;
    const int tile  = blockIdx.x * WAVES_PER_BLOCK + wave;   // 0..8191
    const int b     = tile / TILES_PER_B;
    const int s0    = (tile % TILES_PER_B) * 16;
    const int laneN = lane & 15;          // column / s index within tile
    const int half  = lane >> 4;          // 0: K{0,1}, 1: K{2,3}

    const float* __restrict__ hrow = hidden + (size_t)b * H_DIM;
    const float* __restrict__ erow =
        enc + ((size_t)(s0 + laneN) * B_DIM + (size_t)b) * H_DIM;

    v8f acc = {};  // C = 0

    #pragma unroll 4
    for (int h = 0; h < H_DIM; h += 4) {
        // A fragment: uniform hidden values (hot in cache / scalarized)
        v4f hv = *(const v4f*)(hrow + h);
        v2f a;
        a.x = half ? hv.z : hv.x;
        a.y = half ? hv.w : hv.y;

        // B fragment: streaming, nontemporal (512MB one-shot stream)
        v4f ev = __builtin_nontemporal_load((const v4f*)(erow + h));
        v2f bb;
        bb.x = half ? ev.y : ev.x;   // VGPR0: K0 (lanes 0-15) / K1 (16-31)
        bb.y = half ? ev.w : ev.z;   // VGPR1: K2 (lanes 0-15) / K3 (16-31)

        // D = A*B + C  (8 args: neg_a, A, neg_b, B, c_mod, C, reuse_a, reuse_b)
        acc = __builtin_amdgcn_wmma_f32_16x16x4_f32(
            false, a, false, bb, (short)0, acc, false, false);
    }

    // Every D row is identical; lane n (n<16) column n holds score[s0+n].
    if (lane < 16) {
        scores[(size_t)b * S_DIM + s0 + lane] = acc[0];
    }
}

// ---------------------------------------------------------------------------
// Kernel 2: in-place row softmax over S=2048, one 256-thread block per batch.
// Max-stabilized; wave32 shfl reduction + LDS across the 8 waves.
// ---------------------------------------------------------------------------
__global__ __launch_bounds__(256) void softmax_rows(float* __restrict__ data)
{
    const int b = blockIdx.x;
    const int t = threadIdx.x;
    float* __restrict__ row = data + (size_t)b * S_DIM;

    __shared__ float red[WAVES_PER_BLOCK];

    // Load 8 elements per thread
    float v[8];
    float m = -INFINITY;
    #pragma unroll
    for (int i = 0; i < 8; ++i) {
        v[i] = row[t + i * 256];
        m = fmaxf(m, v[i]);
    }

    // Block max
    #pragma unroll
    for (int off = 16; off > 0; off >>= 1)
        m = fmaxf(m, __shfl_xor(m, off, 32));
    if ((t & 31) == 0) red[t >> 5] = m;
    __syncthreads();
    float rowmax = red[0];
    #pragma unroll
    for (int i = 1; i < WAVES_PER_BLOCK; ++i) rowmax = fmaxf(rowmax, red[i]);
    __syncthreads();

    // Exp + block sum
    float s = 0.0f;
    #pragma unroll
    for (int i = 0; i < 8; ++i) {
        v[i] = __expf(v[i] - rowmax);
        s += v[i];
    }
    #pragma unroll
    for (int off = 16; off > 0; off >>= 1)
        s += __shfl_xor(s, off, 32);
    if ((t & 31) == 0) red[t >> 5] = s;
    __syncthreads();
    float total = 0.0f;
    #pragma unroll
    for (int i = 0; i < WAVES_PER_BLOCK; ++i) total += red[i];

    const float inv = 1.0f / total;
    #pragma unroll
    for (int i = 0; i < 8; ++i)
        row[t + i * 256] = v[i] * inv;
}

// ---------------------------------------------------------------------------
// Launch: inputs per setup_inputs order: d_in[0]=hidden [1,B,H], d_in[1]=enc
// [S,B,H]. d_out = [B,1,S] floats (== B*S) — used directly as score buffer.
// ---------------------------------------------------------------------------
extern "C" void kernel_launch(void* const* d_in, const int* in_sizes, int n_in,
                              void* d_out, int out_size, void* d_ws, size_t ws_size,
                              hipStream_t stream) {
    (void)in_sizes; (void)n_in; (void)out_size; (void)d_ws; (void)ws_size;

    const float* hidden = (const float*)d_in[0];
    const float* enc    = (const float*)d_in[1];
    float*       out    = (float*)d_out;

    const int n_tiles = B_DIM * TILES_PER_B;                 // 8192
    const int blocks1 = n_tiles / WAVES_PER_BLOCK;           // 1024

    attn_scores_wmma<<<blocks1, 256, 0, stream>>>(hidden, enc, out);
    softmax_rows<<<B_DIM, 256, 0, stream>>>(out);
}